// Marginal_loss_43997644980380
// MI455X (gfx1250) — compile-verified
//
#include <hip/hip_runtime.h>
#include <hip/hip_bf16.h>
#include <math.h>

#define GLOBAL_AS __attribute__((address_space(1)))
#define LDS_AS    __attribute__((address_space(3)))

typedef int v4i __attribute__((__vector_size__(4 * sizeof(int))));

#ifndef __has_builtin
#define __has_builtin(x) 0
#endif

// ---- CDNA5 async global->LDS copy (128b per lane) -------------------------
__device__ __forceinline__ void async_copy16(const float* gsrc, float* ldst) {
#if __has_builtin(__builtin_amdgcn_global_load_async_to_lds_b128)
    __builtin_amdgcn_global_load_async_to_lds_b128(
        (GLOBAL_AS v4i*)gsrc, (LDS_AS v4i*)ldst, 0, 0);
#else
    unsigned lds_off = (unsigned)(unsigned long long)(LDS_AS void*)ldst;
    asm volatile("global_load_async_to_lds_b128 %0, %1, off"
                 :: "v"(lds_off), "v"(gsrc) : "memory");
#endif
}

template <int N>
__device__ __forceinline__ void wait_asynccnt() {
#if __has_builtin(__builtin_amdgcn_s_wait_asynccnt)
    __builtin_amdgcn_s_wait_asynccnt(N);
#else
    if (N == 0) asm volatile("s_wait_asynccnt 0x0" ::: "memory");
    else        asm volatile("s_wait_asynccnt 0x1" ::: "memory");
#endif
}

__device__ __forceinline__ float wave32_sum(float v) {
#pragma unroll
    for (int m = 16; m > 0; m >>= 1) v += __shfl_xor(v, m, 32);
    return v;
}

// ---- Kernel A: per-batch-row margin contribution --------------------------
// One block (256 thr = 8 wave32) per batch row i. Streams the 10 gloss rows
// (4KB each) through a double-buffered async LDS pipeline.
__global__ __launch_bounds__(256) void margin_kernel(
    const float* __restrict__ sentence, const float* __restrict__ all_gloss,
    const int* __restrict__ sense_mask, const int* __restrict__ sense_ids,
    float* __restrict__ ws_margin) {
    __shared__ __align__(16) float buf[2][1024];
    __shared__ float red2[10][8];
    __shared__ float dvals[10];

    const int tid  = threadIdx.x;
    const int i    = blockIdx.x;
    const int lane = tid & 31;
    const int wave = tid >> 5;

    const float* sent  = sentence + (size_t)i * 1024;
    const float* gbase = all_gloss + (size_t)i * (10 * 1024);

    // sentence row held in registers: thread t owns elements [4t, 4t+4)
    const float4 sv = reinterpret_cast<const float4*>(sent)[tid];

    // prologue: kick off sense 0
    async_copy16(gbase + tid * 4, &buf[0][tid * 4]);

    for (int s = 0; s < 10; ++s) {
        const int cur = s & 1;
        if (s < 9) {
            // prefetch next sense into the other buffer (consumed last iter,
            // protected by the barrier at the end of the previous iteration)
            async_copy16(gbase + (s + 1) * 1024 + tid * 4, &buf[cur ^ 1][tid * 4]);
            wait_asynccnt<1>();   // oldest (current buffer) done, prefetch in flight
        } else {
            wait_asynccnt<0>();
        }
        __syncthreads();

        const float4 g = reinterpret_cast<const float4*>(&buf[cur][0])[tid];
        const float dx = sv.x - g.x + 1e-6f;
        const float dy = sv.y - g.y + 1e-6f;
        const float dz = sv.z - g.z + 1e-6f;
        const float dw = sv.w - g.w + 1e-6f;
        float p = dx * dx + dy * dy + dz * dz + dw * dw;

        p = wave32_sum(p);
        if (lane == 0) red2[s][wave] = p;
        __syncthreads();   // also fences buffer reuse before next prefetch
    }

    if (tid < 10) {
        float t = 0.f;
#pragma unroll
        for (int w = 0; w < 8; ++w) t += red2[tid][w];
        dvals[tid] = sqrtf(t);
    }
    __syncthreads();

    if (tid == 0) {
        const int sid = sense_ids[i];
        const float pos = dvals[sid];
        float minv = 3.402823466e38f;
        int   mini = 0;
#pragma unroll
        for (int s = 0; s < 10; ++s) {
            const bool valid = (sense_mask[i * 10 + s] != 0) && (s != sid);
            const float m = valid ? dvals[s] : 10.0f;   // SENTINEL in sqrt domain
            if (m < minv) { minv = m; mini = s; }        // strict < => first tie
        }
        const float neg = dvals[mini];                   // UNMASKED distance
        const float c = pos - neg + 0.5f;                // MARGIN_EPS
        ws_margin[i] = fmaxf(c, 0.0f);
    }
}

// ---- Kernel B: alpha term partials ---------------------------------------
__global__ __launch_bounds__(256) void alpha_kernel(
    const float* __restrict__ alpha, float* __restrict__ ws_alpha, int nrows) {
    __shared__ float tile[2560];
    __shared__ float red8[8];
    const int tid  = threadIdx.x;
    const int b    = blockIdx.x;
    const int base = b * 2560;
    const int total = nrows * 10;

    for (int k = tid; k < 2560; k += 256) {
        const int g = base + k;
        tile[k] = (g < total) ? alpha[g] : 0.0f;
    }
    __syncthreads();

    float val = 0.f;
    const int row = b * 256 + tid;
    if (row < nrows) {
        float ssum = 0.f, mx = -3.402823466e38f;
#pragma unroll
        for (int j = 0; j < 10; ++j) {
            const float a = tile[tid * 10 + j];
            ssum += a;
            mx = fmaxf(mx, a);
        }
        // row_sum == 0 -> tmp = ones -> |max-1| = 0
        val = (ssum == 0.0f) ? 0.0f : fabsf(mx - 1.0f);
    }

    val = wave32_sum(val);
    if ((tid & 31) == 0) red8[tid >> 5] = val;
    __syncthreads();
    if (tid < 32) {
        float r = (tid < 8) ? red8[tid] : 0.f;
#pragma unroll
        for (int m = 4; m > 0; m >>= 1) r += __shfl_xor(r, m, 32);
        if (tid == 0) ws_alpha[b] = r;
    }
}

// ---- Kernel C: deterministic final reduction + combine -------------------
__global__ __launch_bounds__(256) void finalize_kernel(
    const float* __restrict__ ws_margin, const float* __restrict__ ws_alpha,
    float* __restrict__ out) {
    __shared__ float red8[8];
    const int tid = threadIdx.x;

    float m = 0.f;
    for (int k = tid; k < 8192; k += 256) m += ws_margin[k];
    m = wave32_sum(m);
    if ((tid & 31) == 0) red8[tid >> 5] = m;
    __syncthreads();
    float msum = (tid < 8) ? red8[tid] : 0.f;
    if (tid < 32) {
#pragma unroll
        for (int mm = 4; mm > 0; mm >>= 1) msum += __shfl_xor(msum, mm, 32);
    }
    __syncthreads();

    float a = (tid < 64) ? ws_alpha[tid] : 0.f;
    a = wave32_sum(a);
    if ((tid & 31) == 0) red8[tid >> 5] = a;
    __syncthreads();
    float asum = (tid < 8) ? red8[tid] : 0.f;
    if (tid < 32) {
#pragma unroll
        for (int mm = 4; mm > 0; mm >>= 1) asum += __shfl_xor(asum, mm, 32);
    }

    if (tid == 0) {
        out[0] = msum * 0.875f + asum * 0.125f;  // loss (train mode)
        out[1] = msum;                            // margin
        out[2] = asum;                            // alpha_term
    }
}

extern "C" void kernel_launch(void* const* d_in, const int* in_sizes, int n_in,
                              void* d_out, int out_size, void* d_ws, size_t ws_size,
                              hipStream_t stream) {
    (void)in_sizes; (void)n_in; (void)out_size; (void)ws_size;
    const float* sentence  = (const float*)d_in[0];  // (8192, 1024)
    const float* all_gloss = (const float*)d_in[1];  // (8192, 10, 1024)
    const float* alpha     = (const float*)d_in[2];  // (2700, 6, 10)
    const int*   sense_mask = (const int*)d_in[3];   // (8192, 10)
    const int*   sense_ids  = (const int*)d_in[4];   // (8192,)
    float* out = (float*)d_out;                      // [loss, margin, alpha_term]

    float* ws_margin = (float*)d_ws;         // 8192 floats
    float* ws_alpha  = ws_margin + 8192;     // 64 floats

    margin_kernel<<<8192, 256, 0, stream>>>(sentence, all_gloss, sense_mask,
                                            sense_ids, ws_margin);
    alpha_kernel<<<64, 256, 0, stream>>>(alpha, ws_alpha, 16200);
    finalize_kernel<<<1, 256, 0, stream>>>(ws_margin, ws_alpha, out);
}